// ResidualBlock_4037269259025
// MI455X (gfx1250) — compile-verified
//
#include <hip/hip_runtime.h>

#define NN 100000
#define NE 1600000
#define DIM 128
#define NG 64
#define LN_EPS 1e-5f
#define LDS_STRIDE 132   // padded row stride (words); float4-aligned, bank-conflict free

typedef __attribute__((ext_vector_type(2))) float v2f;
typedef __attribute__((ext_vector_type(8))) float v8f;

__device__ __forceinline__ void atomic_add_f32(float* p, float v) {
    __hip_atomic_fetch_add(p, v, __ATOMIC_RELAXED, __HIP_MEMORY_SCOPE_AGENT);
}

// ---------------- agg = (1+eps) * x  (float4) ----------------
__global__ void seed_kernel(float4* __restrict__ agg, const float4* __restrict__ x,
                            const float* __restrict__ eps, int n4) {
    int i = blockIdx.x * blockDim.x + threadIdx.x;
    float s = 1.0f + eps[0];
    if (i < n4) {
        float4 v = x[i];
        v.x *= s; v.y *= s; v.z *= s; v.w *= s;
        agg[i] = v;
    }
}

// ---------------- edge scatter: agg[dst] += relu(x[src] + ea) ----------------
__global__ void edge_kernel(float* __restrict__ agg, const float* __restrict__ x,
                            const int* __restrict__ src, const int* __restrict__ dst,
                            const float* __restrict__ ea) {
    int t = blockIdx.x * blockDim.x + threadIdx.x;
    int e = t >> 5;
    int lane = t & 31;
    if (e >= NE) return;
    int s = src[e];
    int d = dst[e];
    const float4 xv = *(const float4*)(x + (size_t)s * DIM + lane * 4);
    const float4 av = *(const float4*)(ea + (size_t)e * DIM + lane * 4);
    float4 m;
    m.x = fmaxf(xv.x + av.x, 0.f);
    m.y = fmaxf(xv.y + av.y, 0.f);
    m.z = fmaxf(xv.z + av.z, 0.f);
    m.w = fmaxf(xv.w + av.w, 0.f);
    float* p = agg + (size_t)d * DIM + lane * 4;
    atomic_add_f32(p + 0, m.x);
    atomic_add_f32(p + 1, m.y);
    atomic_add_f32(p + 2, m.z);
    atomic_add_f32(p + 3, m.w);
}

// ---------------- out = relu(in @ W1 + b1) @ W2 + b2  via V_WMMA_F32_16X16X4_F32 ----------------
// blockDim = 64 (2 waves). Each wave computes 16 rows x 128 cols. No early returns: EXEC
// stays all-ones around every WMMA (required by CDNA5).
__global__ __launch_bounds__(64) void mlp_kernel(
    float* __restrict__ out, const float* __restrict__ in,
    const float* __restrict__ W1, const float* __restrict__ b1,
    const float* __restrict__ W2, const float* __restrict__ b2, int n_rows) {
    __shared__ float sIn[2][16 * LDS_STRIDE];
    __shared__ float sH [2][16 * LDS_STRIDE];
    const int wave = threadIdx.x >> 5;
    const int lane = threadIdx.x & 31;
    const int m  = lane & 15;   // row-in-tile for A; col-in-tile for B/C
    const int kh = lane >> 4;   // which K-half this lane holds (A/B layout)
    const int rowBase = blockIdx.x * 32 + wave * 16;

    // Stage the 16x128 input tile into LDS (coalesced float4 loads).
    for (int i = lane; i < 16 * (DIM / 4); i += 32) {
        int r  = i >> 5;        // 0..15
        int c4 = i & 31;        // float4 column
        int gr = rowBase + r;
        if (gr >= n_rows) gr = n_rows - 1;   // clamp instead of branch: keeps EXEC full
        const float4 v = *(const float4*)(in + (size_t)gr * DIM + c4 * 4);
        *(float4*)&sIn[wave][r * LDS_STRIDE + c4 * 4] = v;
    }
    __syncthreads();

    // GEMM1: sH = relu(sIn @ W1 + b1)
    for (int jt = 0; jt < 8; ++jt) {
        const int col = jt * 16 + m;
        const float bias = b1[col];
        // Per-lane B base: rows 2*kh / 2*kh+1 of W1, column `col`; kk advances by
        // constant byte offsets -> backend emits immediate-offset clause loads.
        const float* wp = W1 + 2 * kh * DIM + col;
        v8f acc = {bias, bias, bias, bias, bias, bias, bias, bias};
        #pragma unroll
        for (int kk = 0; kk < DIM; kk += 4) {
            v2f A = *(const v2f*)&sIn[wave][m * LDS_STRIDE + kk + 2 * kh];
            v2f B;
            B.x = wp[kk * DIM];        // W1[(kk+2kh)  *DIM + col]
            B.y = wp[kk * DIM + DIM];  // W1[(kk+2kh+1)*DIM + col]
            acc = __builtin_amdgcn_wmma_f32_16x16x4_f32(
                false, A, false, B, (short)0, acc, false, false);
        }
        #pragma unroll
        for (int r = 0; r < 8; ++r) {
            float v = acc[r] > 0.f ? acc[r] : 0.f;
            sH[wave][(r + 8 * kh) * LDS_STRIDE + col] = v;   // C/D layout: (M=r+8*kh, N=col)
        }
    }
    __syncthreads();

    // GEMM2: out = sH @ W2 + b2
    for (int jt = 0; jt < 8; ++jt) {
        const int col = jt * 16 + m;
        const float bias = b2[col];
        const float* wp = W2 + 2 * kh * DIM + col;
        v8f acc = {bias, bias, bias, bias, bias, bias, bias, bias};
        #pragma unroll
        for (int kk = 0; kk < DIM; kk += 4) {
            v2f A = *(const v2f*)&sH[wave][m * LDS_STRIDE + kk + 2 * kh];
            v2f B;
            B.x = wp[kk * DIM];
            B.y = wp[kk * DIM + DIM];
            acc = __builtin_amdgcn_wmma_f32_16x16x4_f32(
                false, A, false, B, (short)0, acc, false, false);
        }
        #pragma unroll
        for (int r = 0; r < 8; ++r) {
            int gr = rowBase + r + 8 * kh;
            if (gr < n_rows) out[(size_t)gr * DIM + col] = acc[r];
        }
    }
}

// ---------------- per-graph stats: sum, sumsq, count ----------------
__global__ void zero_stats_kernel(float* __restrict__ s) {
    int i = threadIdx.x;
    if (i < 3 * NG) s[i] = 0.f;
}

__global__ void stats_kernel(const float* __restrict__ h, const int* __restrict__ n2g,
                             float* __restrict__ stats) {
    int t = blockIdx.x * blockDim.x + threadIdx.x;
    int node = t >> 5;
    int lane = t & 31;
    if (node >= NN) return;
    const float4 v = *(const float4*)(h + (size_t)node * DIM + lane * 4);
    float s = v.x + v.y + v.z + v.w;
    float q = v.x * v.x + v.y * v.y + v.z * v.z + v.w * v.w;
    #pragma unroll
    for (int off = 16; off >= 1; off >>= 1) {
        s += __shfl_xor(s, off, 32);
        q += __shfl_xor(q, off, 32);
    }
    if (lane == 0) {
        int g = n2g[node];
        atomic_add_f32(&stats[g], s);
        atomic_add_f32(&stats[NG + g], q);
        atomic_add_f32(&stats[2 * NG + g], (float)DIM);
    }
}

// ---------------- graph layernorm (+relu), float4 ----------------
// mode 0: out = relu(LN(in));  if seed_out != null also write seed_out = (1+eps_next)*out
//         (fuses the next conv's (1+eps)*x seed pass into this one).
// mode 1: out = relu((LN(in) + resid) * 0.5)
__global__ void ln_kernel(float4* __restrict__ out, const float4* __restrict__ in,
                          const int* __restrict__ n2g,
                          const float4* __restrict__ gamma, const float4* __restrict__ beta,
                          const float* __restrict__ stats,
                          const float4* __restrict__ resid, int final_mode,
                          float4* __restrict__ seed_out, const float* __restrict__ eps_next) {
    int i = blockIdx.x * blockDim.x + threadIdx.x;     // float4 index
    if (i >= NN * (DIM / 4)) return;
    int node = i >> 5;                                  // 32 float4 per row
    int d4 = i & 31;
    int g = n2g[node];
    float c = fmaxf(stats[2 * NG + g], 1.0f);
    float mean = stats[g] / c;
    float var = stats[NG + g] / c - mean * mean;
    float inv = rsqrtf(var + LN_EPS);
    float4 v = in[i];
    float4 gm = gamma[d4];
    float4 bt = beta[d4];
    v.x = (v.x - mean) * inv * gm.x + bt.x;
    v.y = (v.y - mean) * inv * gm.y + bt.y;
    v.z = (v.z - mean) * inv * gm.z + bt.z;
    v.w = (v.w - mean) * inv * gm.w + bt.w;
    if (final_mode) {
        float4 r = resid[i];
        v.x = (v.x + r.x) * 0.5f;
        v.y = (v.y + r.y) * 0.5f;
        v.z = (v.z + r.z) * 0.5f;
        v.w = (v.w + r.w) * 0.5f;
    }
    v.x = fmaxf(v.x, 0.f);
    v.y = fmaxf(v.y, 0.f);
    v.z = fmaxf(v.z, 0.f);
    v.w = fmaxf(v.w, 0.f);
    out[i] = v;
    if (seed_out) {
        float s = 1.0f + eps_next[0];
        float4 sv;
        sv.x = v.x * s; sv.y = v.y * s; sv.z = v.z * s; sv.w = v.w * s;
        seed_out[i] = sv;
    }
}

extern "C" void kernel_launch(void* const* d_in, const int* in_sizes, int n_in,
                              void* d_out, int out_size, void* d_ws, size_t ws_size,
                              hipStream_t stream) {
    const float* x    = (const float*)d_in[0];
    const int*   ei   = (const int*)d_in[1];
    const float* ea   = (const float*)d_in[2];
    const int*   n2g  = (const int*)d_in[3];
    const float* eps1 = (const float*)d_in[4];
    const float* eps2 = (const float*)d_in[5];
    const float* W1a = (const float*)d_in[6];
    const float* b1a = (const float*)d_in[7];
    const float* W2a = (const float*)d_in[8];
    const float* b2a = (const float*)d_in[9];
    const float* W1b = (const float*)d_in[10];
    const float* b1b = (const float*)d_in[11];
    const float* W2b = (const float*)d_in[12];
    const float* b2b = (const float*)d_in[13];
    const float* g1  = (const float*)d_in[14];
    const float* bt1 = (const float*)d_in[15];
    const float* g2  = (const float*)d_in[16];
    const float* bt2 = (const float*)d_in[17];
    float* out = (float*)d_out;

    float* bufA  = (float*)d_ws;                      // N*D
    float* bufB  = bufA + (size_t)NN * DIM;           // N*D
    float* stats = bufB + (size_t)NN * DIM;           // 3*NG

    const int* src = ei;
    const int* dst = ei + NE;

    const int ND4     = NN * (DIM / 4);
    const int gridND4 = (ND4 + 255) / 256;
    const int gridE   = (NE * 32 + 255) / 256;        // one wave per edge
    const int gridNW  = (NN * 32 + 255) / 256;        // one wave per node
    const int gridMlp = (NN + 31) / 32;               // 32 rows per block

    // ---- conv1 ----
    seed_kernel<<<gridND4, 256, 0, stream>>>((float4*)bufA, (const float4*)x, eps1, ND4);
    edge_kernel<<<gridE, 256, 0, stream>>>(bufA, x, src, dst, ea);
    mlp_kernel<<<gridMlp, 64, 0, stream>>>(bufB, bufA, W1a, b1a, W2a, b2a, NN);
    zero_stats_kernel<<<1, 256, 0, stream>>>(stats);
    stats_kernel<<<gridNW, 256, 0, stream>>>(bufB, n2g, stats);
    // LN1 writes y1 into bufA and, fused, the conv2 seed (1+eps2)*y1 into bufB.
    ln_kernel<<<gridND4, 256, 0, stream>>>((float4*)bufA, (const float4*)bufB, n2g,
                                           (const float4*)g1, (const float4*)bt1,
                                           stats, nullptr, 0, (float4*)bufB, eps2);

    // ---- conv2 ----  (bufA = y1, bufB = (1+eps2)*y1 already seeded)
    edge_kernel<<<gridE, 256, 0, stream>>>(bufB, bufA, src, dst, ea);
    mlp_kernel<<<gridMlp, 64, 0, stream>>>(out, bufB, W1b, b1b, W2b, b2b, NN);
    zero_stats_kernel<<<1, 256, 0, stream>>>(stats);
    stats_kernel<<<gridNW, 256, 0, stream>>>(out, n2g, stats);
    ln_kernel<<<gridND4, 256, 0, stream>>>((float4*)out, (const float4*)out, n2g,
                                           (const float4*)g2, (const float4*)bt2,
                                           stats, (const float4*)x, 1, nullptr, nullptr);
}